// QuadModel_19542101197062
// MI455X (gfx1250) — compile-verified
//
#include <hip/hip_runtime.h>
#include <float.h>

typedef __attribute__((ext_vector_type(2))) float v2f;
typedef __attribute__((ext_vector_type(8))) float v8f;

#define SLEN 8192
#define DDIM 768
#define NA 256
#define NO 256
#define MAXW 40
#define NPAIR (NA * NO)
#define NC 32           // padded classifier column count (2 valid + 13 cat + 3 pol + pad)

// workspace layout (in floats)
#define WS_ASP 0                        // 256 x 768
#define WS_OPI (WS_ASP + NA * DDIM)     // 256 x 768
#define WS_W   (WS_OPI + NO * DDIM)     // 1536 x 32 packed weights (top half rows 0..767 = asp side)
#define WS_B   (WS_W + 2 * DDIM * NC)   // 32 packed bias
#define WS_PA  (WS_B + NC)              // 256 x 32  asp partial scores
#define WS_PO  (WS_PA + NA * NC)        // 256 x 32  opi partial scores

// ---------------------------------------------------------------------------
// Kernel 0: pack W_valid|W_cat|W_pol into a zero-padded (1536 x 32) matrix,
// and the biases into a 32-vector. Cols: 0-1 valid, 2-14 cat, 15-17 pol.
// ---------------------------------------------------------------------------
__global__ void pack_wb_kernel(const float* __restrict__ Wv, const float* __restrict__ bv,
                               const float* __restrict__ Wc, const float* __restrict__ bc,
                               const float* __restrict__ Wp, const float* __restrict__ bp,
                               float* __restrict__ ws) {
    int idx = blockIdx.x * 256 + threadIdx.x;
    if (idx < 2 * DDIM * NC) {
        int row = idx / NC, col = idx % NC;
        float v = 0.0f;
        if (col < 2)       v = Wv[row * 2 + col];
        else if (col < 15) v = Wc[row * 13 + (col - 2)];
        else if (col < 18) v = Wp[row * 3 + (col - 15)];
        ws[WS_W + idx] = v;
    }
    if (idx < NC) {
        float v = 0.0f;
        if (idx < 2)       v = bv[idx];
        else if (idx < 15) v = bc[idx - 2];
        else if (idx < 18) v = bp[idx - 15];
        ws[WS_B + idx] = v;
    }
}

// ---------------------------------------------------------------------------
// Kernel 1: span max-pool. One block per span (256 aspects then 256 opinions),
// 256 threads, 3 feature dims per thread. Span bounds are block-uniform.
// ---------------------------------------------------------------------------
__global__ void span_maxpool_kernel(const float* __restrict__ word_rep,
                                    const int* __restrict__ aspects,
                                    const int* __restrict__ opinions,
                                    float* __restrict__ ws) {
    int b = blockIdx.x;
    const int* span = (b < NA) ? (aspects + 2 * b) : (opinions + 2 * (b - NA));
    float* out = ws + ((b < NA) ? (WS_ASP + b * DDIM) : (WS_OPI + (b - NA) * DDIM));
    int head = span[0], tail = span[1];
    int t = threadIdx.x;
    float m0 = -FLT_MAX, m1 = -FLT_MAX, m2 = -FLT_MAX;
    for (int w = 0; w < MAXW; ++w) {
        int pos = head + w;
        if (pos >= tail) break;               // uniform across block
        const float* row = word_rep + (size_t)pos * DDIM;
        m0 = fmaxf(m0, row[t]);
        m1 = fmaxf(m1, row[t + 256]);
        m2 = fmaxf(m2, row[t + 512]);
    }
    out[t] = m0; out[t + 256] = m1; out[t + 512] = m2;
}

// ---------------------------------------------------------------------------
// Kernel 2: partial-score GEMM via V_WMMA_F32_16X16X4_F32.
//   P_asp = asp_rep(256x768) @ Wpack[0:768, 0:32]
//   P_opi = opi_rep(256x768) @ Wpack[768:1536, 0:32]
// grid = (Mtiles=16, Ntiles=2, side=2), one wave32 per 16x16 output tile.
// Per-lane layouts per ISA 7.12.2 (32-bit A 16x4 / B 4x16 / C 16x16).
// ---------------------------------------------------------------------------
__global__ void part_gemm_kernel(float* __restrict__ ws) {
    int lane = threadIdx.x;                 // 0..31, EXEC all ones
    int m0   = blockIdx.x * 16;
    int n0   = blockIdx.y * 16;
    int side = blockIdx.z;                  // 0 = aspects, 1 = opinions
    const float* rep = ws + (side ? WS_OPI : WS_ASP);
    const float* W   = ws + WS_W + (side ? DDIM * NC : 0);
    float* P         = ws + (side ? WS_PO : WS_PA);

    int r  = lane & 15;                     // row (A) / column (B,D) within tile
    int kh = lane >> 4;                     // K half-select

    v8f c = {};
    for (int kb = 0; kb < DDIM; kb += 4) {
        v2f a, b;
        const float* ap = rep + (size_t)(m0 + r) * DDIM + kb + 2 * kh;
        a.x = ap[0];                                    // K = kb + 2*kh
        a.y = ap[1];                                    // K = kb + 2*kh + 1
        b.x = W[(size_t)(kb + 2 * kh + 0) * NC + n0 + r];
        b.y = W[(size_t)(kb + 2 * kh + 1) * NC + n0 + r];
        c = __builtin_amdgcn_wmma_f32_16x16x4_f32(
                /*neg_a=*/false, a, /*neg_b=*/false, b,
                /*c_mod=*/(short)0, c, /*reuse_a=*/false, /*reuse_b=*/false);
    }
#pragma unroll
    for (int v = 0; v < 8; ++v)
        P[(size_t)(m0 + v + 8 * kh) * NC + n0 + r] = c[v];
}

// ---------------------------------------------------------------------------
// Kernel 3: pair expansion + masking + writeback (HBM-store bound, ~408 MB).
// One block per pair, 128 threads, 3 float4 stores/thread for the 1536-f row.
// ---------------------------------------------------------------------------
__global__ void pair_out_kernel(const float* __restrict__ ws, float* __restrict__ out) {
    int p = blockIdx.x;
    int i = p >> 8, j = p & 255;
    const float* PA = ws + WS_PA + (size_t)i * NC;
    const float* PO = ws + WS_PO + (size_t)j * NC;
    const float* bp = ws + WS_B;
    int t = threadIdx.x;

    __shared__ int validS;
    if (t == 0) {
        float s0 = PA[0] + PO[0] + bp[0];
        float s1 = PA[1] + PO[1] + bp[1];
        validS = (s0 > s1) ? 1 : 0;
    }
    __syncthreads();
    int valid = validS;                      // uniform per block

    const float4* asp = (const float4*)(ws + WS_ASP + (size_t)i * DDIM);
    const float4* opi = (const float4*)(ws + WS_OPI + (size_t)j * DDIM);
    float4* row = (float4*)(out + (size_t)p * (2 * DDIM));
    const float4 z = make_float4(0.f, 0.f, 0.f, 0.f);
#pragma unroll
    for (int u = 0; u < 3; ++u) {
        int q = t + u * 128;                 // 0..383 float4 slots (192 asp + 192 opi)
        float4 v = valid ? ((q < 192) ? asp[q] : opi[q - 192]) : z;
        row[q] = v;
    }

    float* O1 = out + (size_t)NPAIR * 2 * DDIM;   // asp_ids
    float* O2 = O1 + NPAIR;                        // opi_ids
    float* O3 = O2 + NPAIR;                        // cate_out 13
    float* O4 = O3 + (size_t)NPAIR * 13;           // polar_out 3
    float* O5 = O4 + (size_t)NPAIR * 3;            // valid_mask
    if (t == 0)       O1[p] = valid ? (float)i : -1.0f;
    else if (t == 1)  O2[p] = valid ? (float)j : -1.0f;
    else if (t < 15) {
        float s = PA[t] + PO[t] + bp[t];           // packed cols 2..14 = categories
        O3[(size_t)p * 13 + (t - 2)] = valid ? s : 0.0f;
    } else if (t < 18) {
        float s = PA[t] + PO[t] + bp[t];           // packed cols 15..17 = polarity
        O4[(size_t)p * 3 + (t - 15)] = valid ? s : 0.0f;
    } else if (t == 18) {
        O5[p] = valid ? 1.0f : 0.0f;
    }
}

// ---------------------------------------------------------------------------
extern "C" void kernel_launch(void* const* d_in, const int* in_sizes, int n_in,
                              void* d_out, int out_size, void* d_ws, size_t ws_size,
                              hipStream_t stream) {
    const float* word_rep = (const float*)d_in[0];
    const int*   aspects  = (const int*)d_in[1];
    const int*   opinions = (const int*)d_in[2];
    const float* Wv = (const float*)d_in[3];
    const float* bv = (const float*)d_in[4];
    const float* Wc = (const float*)d_in[5];
    const float* bc = (const float*)d_in[6];
    const float* Wp = (const float*)d_in[7];
    const float* bp = (const float*)d_in[8];
    float* ws  = (float*)d_ws;
    float* out = (float*)d_out;

    pack_wb_kernel<<<(2 * DDIM * NC + 255) / 256, 256, 0, stream>>>(Wv, bv, Wc, bc, Wp, bp, ws);
    span_maxpool_kernel<<<NA + NO, 256, 0, stream>>>(word_rep, aspects, opinions, ws);
    part_gemm_kernel<<<dim3(16, 2, 2), 32, 0, stream>>>(ws);
    pair_out_kernel<<<NPAIR, 128, 0, stream>>>(ws, out);
}